// GraphVAE_21663815041514
// MI455X (gfx1250) — compile-verified
//
#include <hip/hip_runtime.h>

typedef __attribute__((ext_vector_type(2))) float v2f;
typedef __attribute__((ext_vector_type(8))) float v8f;

#define N_NODES 8192
#define E_EDGES 262144
#define F_IN    512
#define F_HID   512
#define F_LAT   128

// ---------------------------------------------------------------------------
// Degree / normalization
// ---------------------------------------------------------------------------
__global__ void k_init_deg(float* __restrict__ deg) {
    int i = blockIdx.x * blockDim.x + threadIdx.x;
    if (i < N_NODES) deg[i] = 1.0f;               // self-loop contributes 1
}

__global__ void k_accum_deg(const int* __restrict__ col, float* __restrict__ deg) {
    int e = blockIdx.x * blockDim.x + threadIdx.x;
    if (e < E_EDGES) atomicAdd(&deg[col[e]], 1.0f);
}

__global__ void k_rsqrt_deg(float* __restrict__ deg) {
    int i = blockIdx.x * blockDim.x + threadIdx.x;
    if (i < N_NODES) deg[i] = rsqrtf(deg[i]);     // deg >= 1 always
}

// ---------------------------------------------------------------------------
// Generic fp32 WMMA GEMM: C[M,Nn] = A[M,K] @ B[K,Nn], all row-major.
// One wave computes a 32x32 tile via 2x2 grid of 16x16 f32 accumulators.
// Uses V_WMMA_F32_16X16X4_F32. All dims are multiples of 32/4 -> EXEC all-1s.
// ---------------------------------------------------------------------------
__global__ void k_gemm_f32_wmma(const float* __restrict__ A,
                                const float* __restrict__ B,
                                float* __restrict__ C,
                                int M, int Nn, int K) {
    const int lane = threadIdx.x & 31;
    const int wv   = blockIdx.x * (blockDim.x >> 5) + (threadIdx.x >> 5);
    const int tilesN = Nn >> 5;
    const int tm = (wv / tilesN) << 5;
    const int tn = (wv % tilesN) << 5;
    const int hlf = lane >> 4;        // 0: K,K+1   1: K+2,K+3
    const int l16 = lane & 15;

    v8f c00 = {}, c01 = {}, c10 = {}, c11 = {};
    const float* arow0 = A + (size_t)(tm + l16)      * K;
    const float* arow1 = A + (size_t)(tm + 16 + l16) * K;

    for (int k = 0; k < K; k += 4) {
        const int ka = k + 2 * hlf;
        v2f a0 = *(const v2f*)(arow0 + ka);
        v2f a1 = *(const v2f*)(arow1 + ka);
        v2f b0, b1;
        b0.x = B[(size_t)ka       * Nn + tn + l16];
        b0.y = B[(size_t)(ka + 1) * Nn + tn + l16];
        b1.x = B[(size_t)ka       * Nn + tn + 16 + l16];
        b1.y = B[(size_t)(ka + 1) * Nn + tn + 16 + l16];
        c00 = __builtin_amdgcn_wmma_f32_16x16x4_f32(false, a0, false, b0, (short)0, c00, false, false);
        c01 = __builtin_amdgcn_wmma_f32_16x16x4_f32(false, a0, false, b1, (short)0, c01, false, false);
        c10 = __builtin_amdgcn_wmma_f32_16x16x4_f32(false, a1, false, b0, (short)0, c10, false, false);
        c11 = __builtin_amdgcn_wmma_f32_16x16x4_f32(false, a1, false, b1, (short)0, c11, false, false);
    }

#pragma unroll
    for (int v = 0; v < 8; ++v) {
        const int r0 = tm + v + 8 * hlf;        // C/D layout: VGPR v = row v (+8 for hi lanes)
        const int r1 = tm + 16 + v + 8 * hlf;
        C[(size_t)r0 * Nn + tn + l16]      = c00[v];
        C[(size_t)r0 * Nn + tn + 16 + l16] = c01[v];
        C[(size_t)r1 * Nn + tn + l16]      = c10[v];
        C[(size_t)r1 * Nn + tn + 16 + l16] = c11[v];
    }
}

// ---------------------------------------------------------------------------
// Aggregation: self-loop term init, edge scatter (atomics), bias epilogues
// ---------------------------------------------------------------------------
__global__ void k_selfloop_init(const float* __restrict__ src,
                                const float* __restrict__ dinv,
                                float* __restrict__ dst, int F) {
    size_t i = (size_t)blockIdx.x * blockDim.x + threadIdx.x;
    if (i >= (size_t)N_NODES * F) return;
    const int node = (int)(i / F);
    const float d = dinv[node];
    dst[i] = d * d * src[i];
}

// one thread = one (edge, 4-feature chunk); coalesced src reads + dst atomics
__global__ void k_edge_scatter(const int* __restrict__ row,
                               const int* __restrict__ col,
                               const float* __restrict__ dinv,
                               const float* __restrict__ src,
                               float* __restrict__ dst, int F4) {
    size_t idx = (size_t)blockIdx.x * blockDim.x + threadIdx.x;
    if (idx >= (size_t)E_EDGES * F4) return;
    const int e = (int)(idx / F4);
    const int c = (int)(idx % F4);
    const int r  = row[e];
    const int cl = col[e];
    const float w = dinv[r] * dinv[cl];
    const float4 s = ((const float4*)(src + (size_t)r * (F4 * 4)))[c];
    float* d = dst + (size_t)cl * (F4 * 4) + c * 4;
    atomicAdd(d + 0, w * s.x);
    atomicAdd(d + 1, w * s.y);
    atomicAdd(d + 2, w * s.z);
    atomicAdd(d + 3, w * s.w);
}

__global__ void k_bias_relu(float* __restrict__ h, const float* __restrict__ b, int F) {
    size_t i = (size_t)blockIdx.x * blockDim.x + threadIdx.x;
    if (i >= (size_t)N_NODES * F) return;
    h[i] = fmaxf(h[i] + b[i & (F - 1)], 0.0f);
}

__global__ void k_bias_copy(float* __restrict__ out, const float* __restrict__ b,
                            int F, float* __restrict__ copy_dst) {
    size_t i = (size_t)blockIdx.x * blockDim.x + threadIdx.x;
    if (i >= (size_t)N_NODES * F) return;
    const float v = out[i] + b[i & (F - 1)];
    out[i] = v;
    if (copy_dst) copy_dst[i] = v;
}

// ---------------------------------------------------------------------------
// Decoder: A_pred = sigmoid(Z @ Z^T), Z[N,128]. Same WMMA scheme; B tile is a
// transposed access of Z, which for row-major Z is the SAME contiguous float2
// row load as A. Fused sigmoid; nontemporal stores for the 256MB output.
// ---------------------------------------------------------------------------
__global__ void k_decode_sigmoid(const float* __restrict__ Z, float* __restrict__ Ap) {
    const int lane = threadIdx.x & 31;
    const int wv   = blockIdx.x * (blockDim.x >> 5) + (threadIdx.x >> 5);
    const int tilesN = N_NODES >> 5;      // 256
    const int tm = (wv / tilesN) << 5;
    const int tn = (wv % tilesN) << 5;
    const int hlf = lane >> 4;
    const int l16 = lane & 15;

    v8f c00 = {}, c01 = {}, c10 = {}, c11 = {};
    const float* a0p = Z + (size_t)(tm + l16)      * F_LAT;
    const float* a1p = Z + (size_t)(tm + 16 + l16) * F_LAT;
    const float* b0p = Z + (size_t)(tn + l16)      * F_LAT;   // B[k][n] = Z[n][k]
    const float* b1p = Z + (size_t)(tn + 16 + l16) * F_LAT;

#pragma unroll 4
    for (int k = 0; k < F_LAT; k += 4) {
        const int ka = k + 2 * hlf;
        v2f a0 = *(const v2f*)(a0p + ka);
        v2f a1 = *(const v2f*)(a1p + ka);
        v2f b0 = *(const v2f*)(b0p + ka);
        v2f b1 = *(const v2f*)(b1p + ka);
        c00 = __builtin_amdgcn_wmma_f32_16x16x4_f32(false, a0, false, b0, (short)0, c00, false, false);
        c01 = __builtin_amdgcn_wmma_f32_16x16x4_f32(false, a0, false, b1, (short)0, c01, false, false);
        c10 = __builtin_amdgcn_wmma_f32_16x16x4_f32(false, a1, false, b0, (short)0, c10, false, false);
        c11 = __builtin_amdgcn_wmma_f32_16x16x4_f32(false, a1, false, b1, (short)0, c11, false, false);
    }

#pragma unroll
    for (int v = 0; v < 8; ++v) {
        const int r0 = tm + v + 8 * hlf;
        const int r1 = tm + 16 + v + 8 * hlf;
        __builtin_nontemporal_store(1.0f / (1.0f + __expf(-c00[v])), Ap + (size_t)r0 * N_NODES + tn + l16);
        __builtin_nontemporal_store(1.0f / (1.0f + __expf(-c01[v])), Ap + (size_t)r0 * N_NODES + tn + 16 + l16);
        __builtin_nontemporal_store(1.0f / (1.0f + __expf(-c10[v])), Ap + (size_t)r1 * N_NODES + tn + l16);
        __builtin_nontemporal_store(1.0f / (1.0f + __expf(-c11[v])), Ap + (size_t)r1 * N_NODES + tn + 16 + l16);
    }
}

// ---------------------------------------------------------------------------
extern "C" void kernel_launch(void* const* d_in, const int* in_sizes, int n_in,
                              void* d_out, int out_size, void* d_ws, size_t ws_size,
                              hipStream_t stream) {
    const int*   ei  = (const int*)  d_in[0];   // [2,E]: row then col
    const float* x   = (const float*)d_in[1];   // [N,512]
    const float* W1  = (const float*)d_in[2];   // [512,512]
    const float* b1  = (const float*)d_in[3];   // [512]
    const float* Wmu = (const float*)d_in[4];   // [512,128]
    const float* bmu = (const float*)d_in[5];   // [128]
    const float* Wlv = (const float*)d_in[6];   // [512,128]
    const float* blv = (const float*)d_in[7];   // [128]
    (void)in_sizes; (void)n_in; (void)out_size; (void)ws_size;

    const int* row = ei;
    const int* col = ei + E_EDGES;

    float* out = (float*)d_out;
    float* Ap     = out;                                   // [N,N]
    float* mu_out = out + (size_t)N_NODES * N_NODES;       // [N,128]
    float* lv_out = mu_out + (size_t)N_NODES * F_LAT;      // [N,128]
    float* z_out  = lv_out + (size_t)N_NODES * F_LAT;      // [N,128]

    // workspace: dinv[8192] | bufA[8192*512] | bufB[8192*512]  (~33.6 MB)
    float* ws   = (float*)d_ws;
    float* dinv = ws;
    float* bufA = ws + N_NODES;                            // xW1, later hWmu/hWlv
    float* bufB = bufA + (size_t)N_NODES * F_HID;          // h
    float* hWmu = bufA;
    float* hWlv = bufA + (size_t)N_NODES * F_LAT;

    const int T = 256;

    // 1) symmetric normalization: dinv_sqrt
    k_init_deg <<<N_NODES / T, T, 0, stream>>>(dinv);
    k_accum_deg<<<E_EDGES / T, T, 0, stream>>>(col, dinv);
    k_rsqrt_deg<<<N_NODES / T, T, 0, stream>>>(dinv);

    // 2) layer 1: xW1 -> aggregate -> relu(+b1)
    {   // 8192x512x512: (256*16) tiles / 8 waves = 512 blocks
        const int tiles = (N_NODES / 32) * (F_HID / 32);
        k_gemm_f32_wmma<<<tiles / 8, T, 0, stream>>>(x, W1, bufA, N_NODES, F_HID, F_IN);
    }
    k_selfloop_init<<<(N_NODES * F_HID) / T, T, 0, stream>>>(bufA, dinv, bufB, F_HID);
    k_edge_scatter <<<(E_EDGES * (F_HID / 4)) / T, T, 0, stream>>>(row, col, dinv, bufA, bufB, F_HID / 4);
    k_bias_relu    <<<(N_NODES * F_HID) / T, T, 0, stream>>>(bufB, b1, F_HID);

    // 3) mu / logvar heads: h@W -> aggregate -> +bias (mu duplicated into z)
    {
        const int tiles = (N_NODES / 32) * (F_LAT / 32);   // 1024 tiles -> 128 blocks
        k_gemm_f32_wmma<<<tiles / 8, T, 0, stream>>>(bufB, Wmu, hWmu, N_NODES, F_LAT, F_HID);
        k_gemm_f32_wmma<<<tiles / 8, T, 0, stream>>>(bufB, Wlv, hWlv, N_NODES, F_LAT, F_HID);
    }
    k_selfloop_init<<<(N_NODES * F_LAT) / T, T, 0, stream>>>(hWmu, dinv, mu_out, F_LAT);
    k_selfloop_init<<<(N_NODES * F_LAT) / T, T, 0, stream>>>(hWlv, dinv, lv_out, F_LAT);
    k_edge_scatter <<<(E_EDGES * (F_LAT / 4)) / T, T, 0, stream>>>(row, col, dinv, hWmu, mu_out, F_LAT / 4);
    k_edge_scatter <<<(E_EDGES * (F_LAT / 4)) / T, T, 0, stream>>>(row, col, dinv, hWlv, lv_out, F_LAT / 4);
    k_bias_copy    <<<(N_NODES * F_LAT) / T, T, 0, stream>>>(mu_out, bmu, F_LAT, z_out);
    k_bias_copy    <<<(N_NODES * F_LAT) / T, T, 0, stream>>>(lv_out, blv, F_LAT, nullptr);

    // 4) decoder: A_pred = sigmoid(z z^T); 65536 tiles -> 8192 blocks
    {
        const int tiles = (N_NODES / 32) * (N_NODES / 32);
        k_decode_sigmoid<<<tiles / 8, T, 0, stream>>>(z_out, Ap);
    }
}